// MoNet_6828998001340
// MI455X (gfx1250) — compile-verified
//
#include <hip/hip_runtime.h>
#include <hip/hip_bf16.h>
#include <math.h>

// ---------------------------------------------------------------------------
// MoNet / GMMConv x4 for gfx1250 (MI455X, wave32).
//   - Edge path (hot): one pass per layer, fp32, b96/b128 vector loads,
//     LDS-staged params, native f32 global atomics for the mean-scatter.
//   - Node path: dense h@root via V_WMMA_F32_16X16X4_F32, fused with
//     mean-divide + bias + ELU. FULL-tile specialization (N%16==0) keeps the
//     fragment loads branch-free so EXEC stays all-ones around the WMMAs.
// ---------------------------------------------------------------------------

typedef __attribute__((ext_vector_type(2))) float v2f;
typedef __attribute__((ext_vector_type(8))) float v8f;

struct __align__(4) F3 { float x, y, z; };

__device__ __forceinline__ void atomic_add_f32(float* p, float v) {
    // Native global_atomic_add_f32 (no CAS loop).
    unsafeAtomicAdd(p, v);
}

// --------------------------------- fills -----------------------------------
__global__ void __launch_bounds__(256) k_fill_zero(float* __restrict__ p, int n) {
    int i = blockIdx.x * blockDim.x + threadIdx.x;
    if (i < n) p[i] = 0.0f;
}

// ------------------------------ degree count -------------------------------
__global__ void __launch_bounds__(256) k_count(const int* __restrict__ dst,
                                               float* __restrict__ cnt, int E) {
    int e = blockIdx.x * blockDim.x + threadIdx.x;
    if (e < E) atomic_add_f32(&cnt[dst[e]], 1.0f);
}

__global__ void __launch_bounds__(256) k_inv_cnt(const float* __restrict__ cnt,
                                                 float* __restrict__ inv, int n) {
    int i = blockIdx.x * blockDim.x + threadIdx.x;
    if (i < n) inv[i] = 1.0f / fmaxf(cnt[i], 1.0f);
}

// ------------------------------- edge kernel -------------------------------
// msg[e,c] = sum_k gauss[e,k] * (x[src[e],:] @ g[:, k*COUT+c]); atomic-add to agg[dst].
template <int CIN, int COUT>
__global__ void __launch_bounds__(256) k_edge(const float* __restrict__ h,
                                              const int* __restrict__ src,
                                              const int* __restrict__ dst,
                                              const float* __restrict__ ea,
                                              const float* __restrict__ g,     // [CIN, 3*COUT]
                                              const float* __restrict__ mu,    // [3,3]
                                              const float* __restrict__ sigma, // [3,3]
                                              float* __restrict__ agg,         // [N, COUT]
                                              int E) {
    constexpr int KK = 3, DD = 3;
    __shared__ float s_g[CIN * KK * COUT];
    __shared__ float s_mu[KK * DD];
    __shared__ float s_is2[KK * DD];
    for (int i = threadIdx.x; i < CIN * KK * COUT; i += blockDim.x) s_g[i] = g[i];
    if (threadIdx.x < KK * DD) {
        s_mu[threadIdx.x] = mu[threadIdx.x];
        float s = sigma[threadIdx.x];
        s_is2[threadIdx.x] = 1.0f / (1e-15f + s * s);
    }
    __syncthreads();

    int e = blockIdx.x * blockDim.x + threadIdx.x;
    if (e >= E) return;

    const int s = src[e];
    const int d = dst[e];

    // Single b96 load of the 12-byte pseudo-coordinate row.
    const F3 av = *reinterpret_cast<const F3*>(ea + (size_t)3 * e);

    float w[KK];
#pragma unroll
    for (int k = 0; k < KK; k++) {
        float d0 = av.x - s_mu[3 * k + 0];
        float d1 = av.y - s_mu[3 * k + 1];
        float d2 = av.z - s_mu[3 * k + 2];
        w[k] = __expf(-0.5f * (d0 * d0 * s_is2[3 * k + 0] +
                               d1 * d1 * s_is2[3 * k + 1] +
                               d2 * d2 * s_is2[3 * k + 2]));
    }

    // Gather the source feature row with b128 loads when possible
    // (workspace rows are 32/64-byte aligned for CIN=8/16).
    float xs[CIN];
    const float* __restrict__ hrow = h + (size_t)s * CIN;
    if constexpr (CIN % 4 == 0) {
#pragma unroll
        for (int i = 0; i < CIN; i += 4) {
            float4 v = *reinterpret_cast<const float4*>(hrow + i);
            xs[i + 0] = v.x; xs[i + 1] = v.y; xs[i + 2] = v.z; xs[i + 3] = v.w;
        }
    } else {
#pragma unroll
        for (int i = 0; i < CIN; i++) xs[i] = hrow[i];
    }

    float* __restrict__ arow = agg + (size_t)d * COUT;
#pragma unroll
    for (int c = 0; c < COUT; c++) {
        float acc = 0.0f;
#pragma unroll
        for (int k = 0; k < KK; k++) {
            float t = 0.0f;
#pragma unroll
            for (int i = 0; i < CIN; i++)
                t = fmaf(xs[i], s_g[i * (KK * COUT) + k * COUT + c], t);
            acc = fmaf(w[k], t, acc);
        }
        atomic_add_f32(arow + c, acc);
    }
}

// ------------------------------- node kernel -------------------------------
// out = elu( agg * inv_cnt + h @ root + bias ), root GEMM via WMMA f32 16x16x4.
// One wave32 per 16-node tile. FULL=true (N%16==0): branch-free fragment
// loads; EXEC stays all-ones around WMMA as the ISA requires.
template <int CIN, int COUT, bool ELU, bool FULL>
__global__ void __launch_bounds__(256) k_node(const float* __restrict__ h,     // [N, CIN]
                                              const float* __restrict__ root,  // [CIN, COUT]
                                              const float* __restrict__ bias,  // [COUT]
                                              const float* __restrict__ agg,   // [N, COUT]
                                              const float* __restrict__ inv,   // [N]
                                              float* __restrict__ out,         // [N, COUT]
                                              int N) {
    const int lane = threadIdx.x & 31;
    const int wave = threadIdx.x >> 5;
    const int wavesPerBlock = blockDim.x >> 5;
    const int row0 = (blockIdx.x * wavesPerBlock + wave) * 16;
    if (row0 >= N) return;  // wave-uniform exit

    const int half = lane >> 4;  // A: K-pair select; C: row-half select
    const int m = lane & 15;     // A: row; B/C: column
    const bool col_ok = (COUT == 16) ? true : (m < COUT);

    const int arow = row0 + m;
    const bool arow_ok = FULL ? true : (arow < N);
    const float* __restrict__ hrow = h + (size_t)arow * CIN;

    v8f c = {};
    constexpr int KT = (CIN + 3) / 4;
#pragma unroll
    for (int kt = 0; kt < KT; kt++) {
        const int k0 = kt * 4 + 2 * half;  // this lane's two K indices: k0, k0+1
        v2f a, b;
        if constexpr (CIN % 4 == 0) {
            a.x = arow_ok ? hrow[k0 + 0] : 0.0f;   // adjacent -> global_load_b64
            a.y = arow_ok ? hrow[k0 + 1] : 0.0f;
            b.x = col_ok ? root[(k0 + 0) * COUT + m] : 0.0f;
            b.y = col_ok ? root[(k0 + 1) * COUT + m] : 0.0f;
        } else {
            a.x = (arow_ok && k0 + 0 < CIN) ? hrow[k0 + 0] : 0.0f;
            a.y = (arow_ok && k0 + 1 < CIN) ? hrow[k0 + 1] : 0.0f;
            b.x = (col_ok && k0 + 0 < CIN) ? root[(k0 + 0) * COUT + m] : 0.0f;
            b.y = (col_ok && k0 + 1 < CIN) ? root[(k0 + 1) * COUT + m] : 0.0f;
        }
        // D = A(16x4 f32) x B(4x16 f32) + C(16x16 f32)
        c = __builtin_amdgcn_wmma_f32_16x16x4_f32(false, a, false, b, (short)0, c,
                                                  false, false);
    }

    const float bv = col_ok ? bias[m] : 0.0f;
    const float* __restrict__ aggbase = agg + (size_t)(row0 + 8 * half) * COUT + m;
    float* __restrict__ outbase = out + (size_t)(row0 + 8 * half) * COUT + m;
    const float* __restrict__ invbase = inv + row0 + 8 * half;
#pragma unroll
    for (int v = 0; v < 8; v++) {
        const int row = row0 + v + 8 * half;  // C layout: VGPR v -> M=v (+8 hi half)
        const bool ok = (FULL ? true : (row < N)) && col_ok;
        if (ok) {
            float val = c[v] + aggbase[(size_t)v * COUT] * invbase[v] + bv;
            if (ELU) val = (val > 0.0f) ? val : (__expf(val) - 1.0f);
            outbase[(size_t)v * COUT] = val;
        }
    }
}

// ------------------------------- launcher ----------------------------------
extern "C" void kernel_launch(void* const* d_in, const int* in_sizes, int n_in,
                              void* d_out, int out_size, void* d_ws, size_t ws_size,
                              hipStream_t stream) {
    const float* x  = (const float*)d_in[0];
    const int*  ei  = (const int*)d_in[1];   // [2, E]
    const float* ea = (const float*)d_in[2]; // [E, 3]

    const float* g1 = (const float*)d_in[3];
    const float* mu1 = (const float*)d_in[4];
    const float* sg1 = (const float*)d_in[5];
    const float* r1 = (const float*)d_in[6];
    const float* b1 = (const float*)d_in[7];
    const float* g2 = (const float*)d_in[8];
    const float* mu2 = (const float*)d_in[9];
    const float* sg2 = (const float*)d_in[10];
    const float* r2 = (const float*)d_in[11];
    const float* b2 = (const float*)d_in[12];
    const float* g3 = (const float*)d_in[13];
    const float* mu3 = (const float*)d_in[14];
    const float* sg3 = (const float*)d_in[15];
    const float* r3 = (const float*)d_in[16];
    const float* b3 = (const float*)d_in[17];
    const float* g4 = (const float*)d_in[18];
    const float* mu4 = (const float*)d_in[19];
    const float* sg4 = (const float*)d_in[20];
    const float* r4 = (const float*)d_in[21];
    const float* b4 = (const float*)d_in[22];

    const int N = in_sizes[0] / 3;
    const int E = in_sizes[1] / 2;
    const int* src = ei;
    const int* dst = ei + E;

    // Workspace layout (fp32): agg[N*16] | hA[N*16] | hB[N*16] | cnt[N] | inv[N]
    float* ws  = (float*)d_ws;
    float* agg = ws;
    float* hA  = agg + (size_t)N * 16;
    float* hB  = hA + (size_t)N * 16;
    float* cnt = hB + (size_t)N * 16;
    float* inv = cnt + N;

    const int T = 256;
    dim3 blk(T);
    auto grid1 = [&](int n) { return dim3((n + T - 1) / T); };
    // node kernel: 8 waves/block, 16 rows per wave -> 128 rows/block
    dim3 ngrid((N + 127) / 128);
    const bool full = (N % 16) == 0;

    // Degree (same graph every layer): count once per call.
    k_fill_zero<<<grid1(N), blk, 0, stream>>>(cnt, N);
    k_count<<<grid1(E), blk, 0, stream>>>(dst, cnt, E);
    k_inv_cnt<<<grid1(N), blk, 0, stream>>>(cnt, inv, N);

#define RUN_LAYER(CI, CO, ELU, HIN, G, MU, SG, R, B, HOUT)                              \
    do {                                                                                \
        k_fill_zero<<<grid1(N * (CO)), blk, 0, stream>>>(agg, N * (CO));                \
        k_edge<CI, CO><<<grid1(E), blk, 0, stream>>>(HIN, src, dst, ea, G, MU, SG, agg, E); \
        if (full)                                                                       \
            k_node<CI, CO, ELU, true><<<ngrid, blk, 0, stream>>>(HIN, R, B, agg, inv, HOUT, N); \
        else                                                                            \
            k_node<CI, CO, ELU, false><<<ngrid, blk, 0, stream>>>(HIN, R, B, agg, inv, HOUT, N); \
    } while (0)

    RUN_LAYER(3, 8, true, x, g1, mu1, sg1, r1, b1, hA);
    RUN_LAYER(8, 16, true, hA, g2, mu2, sg2, r2, b2, hB);
    RUN_LAYER(16, 8, true, hB, g3, mu3, sg3, r3, b3, hA);
    RUN_LAYER(8, 4, false, hA, g4, mu4, sg4, r4, b4, (float*)d_out);
#undef RUN_LAYER
}